// ModularAttributeVQA_50148038148786
// MI455X (gfx1250) — compile-verified
//
#include <hip/hip_runtime.h>
#include <hip/hip_bf16.h>

typedef float v2f __attribute__((ext_vector_type(2)));
typedef float v8f __attribute__((ext_vector_type(8)));

static __device__ __forceinline__ v8f wmma_f32(v2f a, v2f b, v8f c) {
    // D = A(16x4) * B(4x16) + C(16x16), all f32, wave32
    return __builtin_amdgcn_wmma_f32_16x16x4_f32(false, a, false, b, (short)0, c, false, false);
}

// ---- async global->LDS path (CDNA5), with safe fallback ----
#if defined(__AMDGCN__) && __has_builtin(__builtin_amdgcn_global_load_async_to_lds_b128)
#define HAVE_ASYNC_LDS 1
#else
#define HAVE_ASYNC_LDS 0
#endif

#if HAVE_ASYNC_LDS
typedef int v4i_vs __attribute__((__vector_size__(4 * sizeof(int))));
typedef __attribute__((address_space(1))) v4i_vs g_v4i;   // global
typedef __attribute__((address_space(3))) v4i_vs l_v4i;   // LDS
#define AS1(p) ((g_v4i*)(p))
#define AS3(p) ((l_v4i*)(p))
static __device__ __forceinline__ void async_cp16(const float* g, float* l) {
    __builtin_amdgcn_global_load_async_to_lds_b128(AS1(g), AS3(l), 0, 0);
}
static __device__ __forceinline__ void wait_async0() {
#if __has_builtin(__builtin_amdgcn_s_wait_asynccnt)
    __builtin_amdgcn_s_wait_asynccnt(0);
#else
    asm volatile("s_wait_asynccnt 0" ::: "memory");
#endif
}
#endif

namespace vqa {
constexpr int NS       = 8192;          // samples
constexpr int IMG      = 3200;          // img feature dim
constexpr int MAXIN    = 8;             // stored slices (only 3 used)
constexpr int FCn      = 512;
constexpr int MROWS    = NS * 3;        // 24576 rows actually needed
constexpr int PAD_ROWS = NS + 7 * 16;   // per-attr buckets padded to mult of 16
constexpr int MAX_TILES = PAD_ROWS / 16; // 519
}

using namespace vqa;

// ---------------- bucketing kernels ----------------

__global__ void k_init(int* __restrict__ perm, int* __restrict__ counts,
                       int* __restrict__ cursor) {
    int i = blockIdx.x * 256 + threadIdx.x;
    if (i < PAD_ROWS) perm[i] = -1;
    if (i < 8) { counts[i] = 0; cursor[i] = 0; }
}

__global__ void k_count(const int* __restrict__ attrs, int* __restrict__ counts) {
    int n = blockIdx.x * 256 + threadIdx.x;
    if (n < NS) atomicAdd(&counts[attrs[n]], 1);
}

__global__ void k_prefix(const int* __restrict__ counts, int* __restrict__ pbase,
                         int* __restrict__ tileAttr) {
    if (blockIdx.x != 0 || threadIdx.x != 0) return;
    int base = 0;
    for (int a = 0; a < 7; ++a) {
        pbase[a] = base;
        base += (counts[a] + 15) & ~15;   // pad each bucket to multiple of 16
    }
    pbase[7] = base;
    for (int t = 0; t < MAX_TILES; ++t) {
        int r = t * 16;
        int aa = -1;
        for (int a = 0; a < 7; ++a)
            if (r >= pbase[a] && r < pbase[a + 1]) aa = a;
        tileAttr[t] = aa;
    }
}

__global__ void k_scatter(const int* __restrict__ attrs, const int* __restrict__ pbase,
                          int* __restrict__ cursor, int* __restrict__ perm) {
    int n = blockIdx.x * 256 + threadIdx.x;
    if (n < NS) {
        int a = attrs[n];
        int pos = pbase[a] + atomicAdd(&cursor[a], 1);
        perm[pos] = n;
    }
}

// ---------------- Stage A: feats = relu(img[:, 0:3, :] @ W_cnn + b_cnn) ----------------
// C tile 64x128 per block, 8 waves each 16x64 (4 WMMA accum tiles), K-step 32.
// Tiles staged into LDS via GLOBAL_LOAD_ASYNC_TO_LDS_B128 (ASYNCcnt-tracked).

__global__ __launch_bounds__(256) void k_stageA(const float* __restrict__ img,
                                                const float* __restrict__ Wc,
                                                const float* __restrict__ bc,
                                                float* __restrict__ feats) {
    constexpr int TM = 64, TN = 128, TK = 32;
    // row strides: 36 floats (144 B) and 132 floats (528 B) -> 16B multiples,
    // and 36*m mod 64 is distinct for m=0..15 (bank-conflict-free A reads).
    __shared__ float As[TM][TK + 4];
    __shared__ float Bs[TK][TN + 4];

    const int bm   = blockIdx.x * TM;
    const int bn   = blockIdx.y * TN;
    const int tid  = threadIdx.x;
    const int wave = tid >> 5;
    const int lane = tid & 31;
    const int wm   = (wave & 3) * 16;   // wave's M offset in tile
    const int wn   = (wave >> 2) * 64;  // wave's N offset in tile
    const int half = lane >> 4;
    const int lm   = lane & 15;

    v8f acc[4] = {};

    // A loader: 64x32 floats, 8 contiguous per thread (2 x 16B chunks)
    const int la_m = tid >> 2;
    const int la_k = (tid & 3) * 8;
    const int r    = bm + la_m;                       // compact row n*3+s
    const long gA  = (long)((r / 3) * MAXIN + (r % 3)) * IMG;
    // B loader: 32x128 floats, 16 contiguous per thread (4 x 16B chunks)
    const int lb_k = tid >> 3;
    const int lb_n = (tid & 7) * 16;

    for (int kb = 0; kb < IMG; kb += TK) {
#if HAVE_ASYNC_LDS
        {
            const float* g0 = img + gA + kb + la_k;
            float*       l0 = &As[la_m][la_k];
            async_cp16(g0,     l0);
            async_cp16(g0 + 4, l0 + 4);
            const float* g1 = Wc + (long)(kb + lb_k) * FCn + bn + lb_n;
            float*       l1 = &Bs[lb_k][lb_n];
            async_cp16(g1,      l1);
            async_cp16(g1 + 4,  l1 + 4);
            async_cp16(g1 + 8,  l1 + 8);
            async_cp16(g1 + 12, l1 + 12);
        }
        wait_async0();
#else
#pragma unroll
        for (int j = 0; j < 8; ++j)
            As[la_m][la_k + j] = img[gA + kb + la_k + j];
#pragma unroll
        for (int j = 0; j < 16; ++j)
            Bs[lb_k][lb_n + j] = Wc[(long)(kb + lb_k) * FCn + bn + lb_n + j];
#endif
        __syncthreads();

#pragma unroll
        for (int kk = 0; kk < TK; kk += 4) {
            v2f av;
            av.x = As[wm + lm][kk + 2 * half + 0];
            av.y = As[wm + lm][kk + 2 * half + 1];
#pragma unroll
            for (int nt = 0; nt < 4; ++nt) {
                v2f bv;
                bv.x = Bs[kk + 2 * half + 0][wn + nt * 16 + lm];
                bv.y = Bs[kk + 2 * half + 1][wn + nt * 16 + lm];
                acc[nt] = wmma_f32(av, bv, acc[nt]);
            }
        }
        __syncthreads();
    }

#pragma unroll
    for (int nt = 0; nt < 4; ++nt) {
        const int col   = bn + wn + nt * 16 + lm;
        const float bias = bc[col];
#pragma unroll
        for (int v = 0; v < 8; ++v) {
            const int row = bm + wm + v + 8 * half;
            float x = acc[nt][v] + bias;
            feats[(long)row * FCn + col] = x > 0.f ? x : 0.f;
        }
    }
}

// ---------------- Stage B: build permuted x rows (plain or product path) ----------------

__global__ __launch_bounds__(256) void k_gather(const float* __restrict__ feats,
                                                const int* __restrict__ perm,
                                                const int* __restrict__ attrs,
                                                float* __restrict__ xg) {
    const int i = blockIdx.x;
    const int n = perm[i];
    float* xrow = xg + (long)i * 1024;
    if (n < 0) {
        for (int c = threadIdx.x; c < 1024; c += 256) xrow[c] = 0.f;
        return;
    }
    const int a = attrs[n];
    const bool dist = (a == 5) || (a == 6);
    const float* f0 = feats + (long)(n * 3 + 0) * FCn;
    const float* f1 = f0 + FCn;
    const float* f2 = f1 + FCn;
    for (int c = threadIdx.x; c < FCn; c += 256) {
        float u0 = f0[c], u1 = f1[c], u2 = f2[c];
        xrow[c]        = dist ? u0 * u1 : u0;
        xrow[c + FCn]  = dist ? u1 * u2 : u1;
    }
}

// ---------------- Stage C: h = relu(x @ W1[a] + b1[a]); scores = h @ W2[a] + b2[a] ----------------
// One 16-row tile per block (all rows share attr a), 8 waves x 64 cols = N 512, K 1024.

__global__ __launch_bounds__(256) void k_stageC(const float* __restrict__ xg,
                                                const float* __restrict__ W1,
                                                const float* __restrict__ b1,
                                                const float* __restrict__ W2,
                                                const float* __restrict__ b2,
                                                const int* __restrict__ perm,
                                                const int* __restrict__ tileAttr,
                                                float* __restrict__ out) {
    constexpr int TK = 32;
    __shared__ float Xs[16][TK + 4];   // 144 B stride
    __shared__ float Hs[16][FCn];

    const int t = blockIdx.x;
    const int a = tileAttr[t];
    if (a < 0) return;

    const int tid  = threadIdx.x;
    const int wave = tid >> 5;
    const int lane = tid & 31;
    const int half = lane >> 4;
    const int lm   = lane & 15;
    const int wn   = wave * 64;

    const float* W1a = W1 + (long)a * 1024 * FCn;

    v8f acc[4] = {};

    // X loader (fallback path): 16x32 floats, 2 per thread
    const int lx_m = tid >> 4;
    const int lx_k = (tid & 15) * 2;
    const long xbase = (long)(t * 16 + lx_m) * 1024;

    for (int kb = 0; kb < 1024; kb += TK) {
#if HAVE_ASYNC_LDS
        if (tid < 128) {                 // 128 x 16B chunks = 16x32 tile
            const int m  = tid >> 3;
            const int kc = (tid & 7) * 4;
            const float* g = xg + (long)(t * 16 + m) * 1024 + kb + kc;
            async_cp16(g, &Xs[m][kc]);
        }
        wait_async0();
#else
        Xs[lx_m][lx_k]     = xg[xbase + kb + lx_k];
        Xs[lx_m][lx_k + 1] = xg[xbase + kb + lx_k + 1];
#endif
        __syncthreads();
#pragma unroll
        for (int kk = 0; kk < TK; kk += 4) {
            v2f av;
            av.x = Xs[lm][kk + 2 * half + 0];
            av.y = Xs[lm][kk + 2 * half + 1];
#pragma unroll
            for (int nt = 0; nt < 4; ++nt) {
                const int col = wn + nt * 16 + lm;
                const long kg = (long)(kb + kk + 2 * half) * FCn + col;
                v2f bv;                        // W1 is hot in L2; read direct
                bv.x = W1a[kg];
                bv.y = W1a[kg + FCn];
                acc[nt] = wmma_f32(av, bv, acc[nt]);
            }
        }
        __syncthreads();
    }

    // h = relu(acc + b1[a]) into LDS
#pragma unroll
    for (int nt = 0; nt < 4; ++nt) {
        const int col   = wn + nt * 16 + lm;
        const float bias = b1[a * FCn + col];
#pragma unroll
        for (int v = 0; v < 8; ++v) {
            float h = acc[nt][v] + bias;
            Hs[v + 8 * half][col] = h > 0.f ? h : 0.f;
        }
    }
    __syncthreads();

    // tiny 512->2 GEMV on wave 0: lane = (row, col) pair
    if (wave == 0) {
        const int row  = lm;
        const int colc = half;     // 0 or 1
        float s = b2[a * 2 + colc];
        const float* W2a = W2 + (long)a * FCn * 2;
        for (int k = 0; k < FCn; ++k)
            s += Hs[row][k] * W2a[k * 2 + colc];
        const int n = perm[t * 16 + row];
        if (n >= 0) out[n * 2 + colc] = s;
    }
}

// ---------------- launcher ----------------

extern "C" void kernel_launch(void* const* d_in, const int* in_sizes, int n_in,
                              void* d_out, int out_size, void* d_ws, size_t ws_size,
                              hipStream_t stream) {
    const float* img   = (const float*)d_in[0];
    const int*   attrs = (const int*)d_in[1];
    const float* Wc    = (const float*)d_in[2];
    const float* bc    = (const float*)d_in[3];
    const float* W1    = (const float*)d_in[4];
    const float* b1    = (const float*)d_in[5];
    const float* W2    = (const float*)d_in[6];
    const float* b2    = (const float*)d_in[7];
    float* out = (float*)d_out;

    // workspace layout (~85 MB)
    float* feats = (float*)d_ws;                              // 24576*512 f32
    float* xg    = feats + (size_t)MROWS * FCn;               // 8304*1024 f32
    int*   perm  = (int*)(xg + (size_t)PAD_ROWS * 1024);      // 8304
    int*   counts   = perm + PAD_ROWS;                        // 8
    int*   cursor   = counts + 8;                             // 8
    int*   pbase    = cursor + 8;                             // 8
    int*   tileAttr = pbase + 8;                              // 519

    k_init   <<<(PAD_ROWS + 255) / 256, 256, 0, stream>>>(perm, counts, cursor);
    k_count  <<<NS / 256, 256, 0, stream>>>(attrs, counts);
    k_prefix <<<1, 1, 0, stream>>>(counts, pbase, tileAttr);
    k_scatter<<<NS / 256, 256, 0, stream>>>(attrs, pbase, cursor, perm);

    dim3 gA(MROWS / 64, FCn / 128);
    k_stageA <<<gA, 256, 0, stream>>>(img, Wc, bc, feats);
    k_gather <<<PAD_ROWS, 256, 0, stream>>>(feats, perm, attrs, xg);
    k_stageC <<<MAX_TILES, 256, 0, stream>>>(xg, W1, b1, W2, b2, perm, tileAttr, out);
}